// PlaneConsistentLoss_73744588472835
// MI455X (gfx1250) — compile-verified
//
#include <hip/hip_runtime.h>

typedef __attribute__((ext_vector_type(2))) float v2f;
typedef __attribute__((ext_vector_type(8))) float v8f;

#define COEF 0.1f
#define NB   1024
#define NT   256
#define NWAVES (NT / 32)

// Reduce 32 per-lane floats to the full wave sum (returned in every lane)
// using V_WMMA_F32_16X16X4_F32 with B = ones.
// A (16x4 f32, 2 VGPRs/lane): VGPR0 holds K=0 (lanes 0-15) / K=2 (lanes 16-31),
// VGPR1 holds K=1 / K=3 -> set VGPR1 = 0 so D[m][n] = acc[m] + acc[m+16].
// Each lane then owns 8 D rows; summing them + shfl_xor(16) gives sum over all
// 16 rows = full 32-lane sum, replicated to every lane.
__device__ __forceinline__ float wave_sum_wmma(float x) {
    v2f a; a[0] = x;    a[1] = 0.0f;
    v2f b; b[0] = 1.0f; b[1] = 1.0f;
    v8f c = {};
    c = __builtin_amdgcn_wmma_f32_16x16x4_f32(
        /*neg_a=*/false, a, /*neg_b=*/false, b,
        /*c_mod=*/(short)0, c, /*reuse_a=*/false, /*reuse_b=*/false);
    float s = ((c[0] + c[1]) + (c[2] + c[3])) + ((c[4] + c[5]) + (c[6] + c[7]));
    s += __shfl_xor(s, 16, 32);
    return s;
}

__device__ __forceinline__ float block_sum(float x, float* lds) {
    float w = wave_sum_wmma(x);
    int lane = threadIdx.x & 31;
    int wid  = threadIdx.x >> 5;
    if (lane == 0) lds[wid] = w;
    __syncthreads();
    float t = 0.0f;
    if (threadIdx.x == 0) {
#pragma unroll
        for (int k = 0; k < NWAVES; ++k) t += lds[k];
    }
    return t;  // valid on thread 0 only
}

__global__ void __launch_bounds__(NT)
plane_loss_partials(const float* __restrict__ xfine,
                    const float* __restrict__ xcoarse,
                    const int*   __restrict__ idx_fine,
                    const int*   __restrict__ idx_coarse,
                    float*       __restrict__ partials,
                    int times) {
    const long long total  = 2LL * times;
    const long long stride = (long long)NB * NT;
    float acc = 0.0f;

    for (long long i = (long long)blockIdx.x * NT + threadIdx.x; i < total; i += stride) {
        const float* x;
        const int4*  idx;
        long long j;
        if (i < times) { x = xfine;   idx = (const int4*)idx_fine;   j = i; }
        else           { x = xcoarse; idx = (const int4*)idx_coarse; j = i - times; }

        int4 q = idx[j];  // global_load_b128 of the 4 indices
        const float* p0 = x + 3LL * q.x;
        const float* p1 = x + 3LL * q.y;
        const float* p2 = x + 3LL * q.z;
        const float* p3 = x + 3LL * q.w;

        float ax = p0[0], ay = p0[1], az = p0[2];
        float e1x = p1[0] - ax, e1y = p1[1] - ay, e1z = p1[2] - az;
        float e2x = p2[0] - ax, e2y = p2[1] - ay, e2z = p2[2] - az;
        float e3x = p3[0] - ax, e3y = p3[1] - ay, e3z = p3[2] - az;

        float cx = e1y * e2z - e1z * e2y;
        float cy = e1z * e2x - e1x * e2z;
        float cz = e1x * e2y - e1y * e2x;
        float vol = cx * e3x + cy * e3y + cz * e3z;
        acc += fabsf(vol);
    }

    __shared__ float lds[NWAVES];
    float t = block_sum(acc, lds);
    if (threadIdx.x == 0) partials[blockIdx.x] = t;
}

__global__ void __launch_bounds__(NT)
plane_loss_finalize(const float* __restrict__ partials,
                    float*       __restrict__ out,
                    int times) {
    float acc = 0.0f;
#pragma unroll
    for (int i = threadIdx.x; i < NB; i += NT) acc += partials[i];

    __shared__ float lds[NWAVES];
    float t = block_sum(acc, lds);
    if (threadIdx.x == 0) out[0] = COEF * t / (float)times;
}

extern "C" void kernel_launch(void* const* d_in, const int* in_sizes, int n_in,
                              void* d_out, int out_size, void* d_ws, size_t ws_size,
                              hipStream_t stream) {
    const float* xfine      = (const float*)d_in[0];
    const float* xcoarse    = (const float*)d_in[1];
    const int*   idx_fine   = (const int*)d_in[2];
    const int*   idx_coarse = (const int*)d_in[3];
    float*       out        = (float*)d_out;
    float*       partials   = (float*)d_ws;   // NB floats = 4 KB scratch

    int times = in_sizes[2] / 4;

    plane_loss_partials<<<NB, NT, 0, stream>>>(xfine, xcoarse, idx_fine, idx_coarse,
                                               partials, times);
    plane_loss_finalize<<<1, NT, 0, stream>>>(partials, out, times);
}